// GNNModel_23665269801228
// MI455X (gfx1250) — compile-verified
//
#include <hip/hip_runtime.h>
#include <hip/hip_bf16.h>

typedef __attribute__((ext_vector_type(2))) float v2f;
typedef __attribute__((ext_vector_type(8))) float v8f;

#define N_NODES 100000
#define F_IN    128
#define F_HID   64
#define F_OUT   32
#define N_EDGES 1600000
#define M_TILES (N_NODES / 16)   // 6250, exact
#define PITCH1  (F_IN + 2)       // 130: padded K-pitch for transposed W1
#define PITCH2  (F_HID + 2)      // 66:  padded K-pitch for transposed W2

// Async global->LDS copy of 16B per lane (CDNA5 VGLOBAL async op, ASYNCcnt).
__device__ __forceinline__ void async_copy_b128(unsigned int ldsByteAddr,
                                                unsigned long long gaddr) {
  asm volatile("global_load_async_to_lds_b128 %0, %1, off"
               :: "v"(ldsByteAddr), "v"(gaddr)
               : "memory");
}
__device__ __forceinline__ void wait_async_and_barrier() {
  asm volatile("s_wait_asynccnt 0" ::: "memory");
  __syncthreads();
}
__device__ __forceinline__ unsigned int lds_off(const void* p) {
  return (unsigned int)(size_t)p;   // generic shared ptr: LDS addr = addr[31:0]
}

// ---------------------------------------------------------------------------
// Kernel 1: xl = x @ W1 + b1  (fp32 WMMA 16x16x4), also writes agg = xl
// (self-loop contribution). One wave per 16-row tile, full 16x64 strip.
// W1 staged once per block: async global->LDS (row-major landing zone), then
// transposed in LDS with padded pitch so each B operand is one ds_load_b64
// landing directly in the even-aligned VGPR pair WMMA consumes.
// WMMA fp32 layouts (wave32):
//   A 16x4 : lane<16 -> row=lane, v={A[row][k],A[row][k+1]}; lane>=16 -> k+2,k+3
//   B 4x16 : lane<16 -> col=lane, v={B[k][col],B[k+1][col]}; lane>=16 -> k+2,k+3
//   C/D    : lane<16 -> col=lane, vgpr r = row r; lane>=16 -> col=lane-16, row 8+r
// ---------------------------------------------------------------------------
__global__ __launch_bounds__(256) void gnn_gemm1_wmma(
    const float* __restrict__ x, const float* __restrict__ W1,
    const float* __restrict__ b1, float* __restrict__ xl,
    float* __restrict__ agg) {
  __shared__ float sStage[F_IN * F_HID];     // 32 KB row-major async landing
  __shared__ float sW1t[F_HID * PITCH1];     // 33.3 KB transposed [col][k]

  {  // cooperative async stage: 2048 x 16B chunks over 256 threads
    const unsigned int base = lds_off(sStage);
    const unsigned long long g = (unsigned long long)(const void*)W1;
    for (int i = threadIdx.x; i < (F_IN * F_HID) / 4; i += 256)
      async_copy_b128(base + (unsigned)i * 16u, g + (unsigned long long)i * 16u);
  }
  wait_async_and_barrier();

  // In-LDS transpose: col fastest -> write stride PITCH1 (odd*2) = 32 distinct
  // banks across a wave32; reads are fully coalesced.
  for (int i = threadIdx.x; i < F_IN * F_HID; i += 256) {
    const int col = i & (F_HID - 1);
    const int k   = i >> 6;
    sW1t[col * PITCH1 + k] = sStage[k * F_HID + col];
  }
  __syncthreads();

  const int lane  = threadIdx.x & 31;
  const int wave  = threadIdx.x >> 5;
  const int mtile = blockIdx.x * 8 + wave;
  if (mtile >= M_TILES) return;              // wave-uniform: EXEC stays all-1s
  const int row   = lane & 15;
  const int khalf = lane >> 4;               // 0 or 1 -> K pair offset
  const int m0    = mtile * 16;

  v8f acc[4] = {};
  const float* aBase = x + (size_t)(m0 + row) * F_IN + khalf * 2;

  for (int k = 0; k < F_IN; k += 4) {
    v2f a = *(const v2f*)(aBase + k);        // A[row][k+2*khalf .. +1]
#pragma unroll
    for (int nt = 0; nt < 4; ++nt) {
      // B[k+2*khalf][col], B[k+2*khalf+1][col]: consecutive -> one ds_load_b64
      v2f b = *(const v2f*)&sW1t[(nt * 16 + row) * PITCH1 + k + khalf * 2];
      acc[nt] = __builtin_amdgcn_wmma_f32_16x16x4_f32(
          false, a, false, b, (short)0, acc[nt], false, false);
    }
  }

#pragma unroll
  for (int nt = 0; nt < 4; ++nt) {
    const float bias = b1[nt * 16 + row];
#pragma unroll
    for (int r = 0; r < 8; ++r) {
      const float  v   = acc[nt][r] + bias;
      const size_t idx = (size_t)(m0 + khalf * 8 + r) * F_HID + nt * 16 + row;
      xl[idx]  = v;                          // gather table for edge messages
      agg[idx] = v;                          // self-loop init of accumulator
    }
  }
}

// ---------------------------------------------------------------------------
// Kernel 2: edge scatter-add. agg[dst] += xl[src], 4 features per thread.
// xl/agg are 25.6 MB each -> resident in the 192 MB L2; atomics RMW in L2.
// Index loads broadcast-coalesce (16 lanes per edge share one 8B segment).
// ---------------------------------------------------------------------------
__global__ __launch_bounds__(256) void gnn_scatter(
    const int* __restrict__ src, const int* __restrict__ dst,
    const float* __restrict__ xl, float* __restrict__ agg) {
  const int t = blockIdx.x * 256 + threadIdx.x;
  if (t >= N_EDGES * 16) return;
  const int e = t >> 4;
  const int g = (t & 15) * 4;                // feature group base
  const int s = src[e];
  const int d = dst[e];
  const float4 v = *(const float4*)(xl + (size_t)s * F_HID + g);
  float* p = agg + (size_t)d * F_HID + g;
  atomicAdd(p + 0, v.x);
  atomicAdd(p + 1, v.y);
  atomicAdd(p + 2, v.z);
  atomicAdd(p + 3, v.w);
}

// ---------------------------------------------------------------------------
// Kernel 3: out = relu(agg) @ W2 + b2  (fp32 WMMA 16x16x4, ReLU fused into
// the A-operand load, bias fused into the store). W2 staged + transposed.
// ---------------------------------------------------------------------------
__global__ __launch_bounds__(256) void gnn_gemm2_wmma(
    const float* __restrict__ agg, const float* __restrict__ W2,
    const float* __restrict__ b2, float* __restrict__ out) {
  __shared__ float sStage[F_HID * F_OUT];    // 8 KB row-major async landing
  __shared__ float sW2t[F_OUT * PITCH2];     // 8.25 KB transposed [col][k]

  {
    const unsigned int base = lds_off(sStage);
    const unsigned long long g = (unsigned long long)(const void*)W2;
    for (int i = threadIdx.x; i < (F_HID * F_OUT) / 4; i += 256)
      async_copy_b128(base + (unsigned)i * 16u, g + (unsigned long long)i * 16u);
  }
  wait_async_and_barrier();

  for (int i = threadIdx.x; i < F_HID * F_OUT; i += 256) {
    const int col = i & (F_OUT - 1);
    const int k   = i >> 5;
    sW2t[col * PITCH2 + k] = sStage[k * F_OUT + col];
  }
  __syncthreads();

  const int lane  = threadIdx.x & 31;
  const int wave  = threadIdx.x >> 5;
  const int mtile = blockIdx.x * 8 + wave;
  if (mtile >= M_TILES) return;              // wave-uniform
  const int row   = lane & 15;
  const int khalf = lane >> 4;
  const int m0    = mtile * 16;

  v8f acc[2] = {};
  const float* aBase = agg + (size_t)(m0 + row) * F_HID + khalf * 2;

  for (int k = 0; k < F_HID; k += 4) {
    v2f a;
    a.x = fmaxf(aBase[k], 0.0f);             // ReLU on the fly
    a.y = fmaxf(aBase[k + 1], 0.0f);
#pragma unroll
    for (int nt = 0; nt < 2; ++nt) {
      v2f b = *(const v2f*)&sW2t[(nt * 16 + row) * PITCH2 + k + khalf * 2];
      acc[nt] = __builtin_amdgcn_wmma_f32_16x16x4_f32(
          false, a, false, b, (short)0, acc[nt], false, false);
    }
  }

#pragma unroll
  for (int nt = 0; nt < 2; ++nt) {
    const float bias = b2[nt * 16 + row];
#pragma unroll
    for (int r = 0; r < 8; ++r) {
      out[(size_t)(m0 + khalf * 8 + r) * F_OUT + nt * 16 + row] =
          acc[nt][r] + bias;
    }
  }
}

// ---------------------------------------------------------------------------
// Launch: gemm1 -> scatter -> gemm2, all on `stream` (graph-capture safe).
// Workspace: xl [N*64 f32] + agg [N*64 f32] = 51.2 MB, fully re-written
// each call before any read (no reliance on prior contents).
// ---------------------------------------------------------------------------
extern "C" void kernel_launch(void* const* d_in, const int* in_sizes, int n_in,
                              void* d_out, int out_size, void* d_ws,
                              size_t ws_size, hipStream_t stream) {
  const float* x  = (const float*)d_in[0];
  const int*   ei = (const int*)d_in[1];   // [2, E]: row 0 = src, row 1 = dst
  const float* W1 = (const float*)d_in[2];
  const float* b1 = (const float*)d_in[3];
  const float* W2 = (const float*)d_in[4];
  const float* b2 = (const float*)d_in[5];
  float* out = (float*)d_out;

  float* xl  = (float*)d_ws;
  float* agg = xl + (size_t)N_NODES * F_HID;

  const dim3 block(256);
  const int  gemmBlocks = (M_TILES + 7) / 8;          // 8 waves (tiles) / block

  gnn_gemm1_wmma<<<gemmBlocks, block, 0, stream>>>(x, W1, b1, xl, agg);

  const int scatterThreads = N_EDGES * 16;            // 25.6M
  gnn_scatter<<<(scatterThreads + 255) / 256, block, 0, stream>>>(
      ei, ei + N_EDGES, xl, agg);

  gnn_gemm2_wmma<<<gemmBlocks, block, 0, stream>>>(agg, W2, b2, out);
}